// LICell_29326036697281
// MI455X (gfx1250) — compile-verified
//
#include <hip/hip_runtime.h>
#include <stdint.h>

// LI cell scan: u[t] = alpha*u[t-1] + (1-alpha)*x[t], alpha = exp(-1/|tau|) per dim.
// Bandwidth-bound (268 MB @ 23.3 TB/s ~= 11.5 us floor). CDNA5 path used:
// GLOBAL_LOAD_ASYNC_TO_LDS_B128 double(quad)-buffered pipeline + s_wait_asynccnt.

#define T_STEPS 128
#define BATCH   256
#define DIM     1024
#define BD      (BATCH * DIM)      // elements per timestep = 262144
#define COLS    (BD / 4)           // float4 columns = 65536
#define TPB     256
#define NBUF    4                  // ring depth (prefetch distance = 3)

__global__ __launch_bounds__(TPB) void li_scan_async(
    const float* __restrict__ x,
    const float* __restrict__ tau,
    float*       __restrict__ out)
{
    __shared__ float4 ring[NBUF][TPB];   // 16 KB

    const int tid = threadIdx.x;
    const int c   = blockIdx.x * TPB + tid;       // float4 column in [0, COLS)
    const int dg  = c & (DIM / 4 - 1);            // dim-group (per-dim params)

    // Per-lane decay constants (computed once; v_exp_f32 path, negligible cost).
    const float4 t4 = reinterpret_cast<const float4*>(tau)[dg];
    float4 al, om;
    al.x = expf(-1.0f / fabsf(t4.x));  om.x = 1.0f - al.x;
    al.y = expf(-1.0f / fabsf(t4.y));  om.y = 1.0f - al.y;
    al.z = expf(-1.0f / fabsf(t4.z));  om.z = 1.0f - al.z;
    al.w = expf(-1.0f / fabsf(t4.w));  om.w = 1.0f - al.w;

    // LDS byte offset of this lane's slot-0 entry. Generic shared pointer's low
    // 32 bits are the LDS offset (per the shared-aperture rules).
    const uint32_t ldsBase =
        (uint32_t)(uintptr_t)(&ring[0][0]) + (uint32_t)tid * 16u;
    // Global byte address of this lane's t=0 element.
    const uint64_t gBase =
        (uint64_t)(uintptr_t)x + (uint64_t)c * 16u;
    const uint64_t gStep = (uint64_t)BD * 4u;     // bytes per timestep (1 MiB)

    // ---- Prologue: launch async loads for t = 0..NBUF-2 -------------------
#pragma unroll
    for (int p = 0; p < NBUF - 1; ++p) {
        uint32_t lo = ldsBase + (uint32_t)p * (TPB * 16u);
        uint64_t ga = gBase + (uint64_t)p * gStep;
        asm volatile("global_load_async_to_lds_b128 %0, %1, off"
                     :: "v"(lo), "v"(ga) : "memory");
    }

    float4 u = make_float4(0.f, 0.f, 0.f, 0.f);
    float4* __restrict__ o4 = reinterpret_cast<float4*>(out);

    // ---- Steady state: issue t+3, wait until load(t) retired, consume t ----
#pragma unroll 1
    for (int t = 0; t < T_STEPS - (NBUF - 1); ++t) {
        const int tp = t + (NBUF - 1);
        uint32_t lo = ldsBase + (uint32_t)(tp & (NBUF - 1)) * (TPB * 16u);
        uint64_t ga = gBase + (uint64_t)tp * gStep;
        asm volatile("global_load_async_to_lds_b128 %0, %1, off"
                     :: "v"(lo), "v"(ga) : "memory");
        // Async loads complete in order: <=3 outstanding => load(t) is done.
        asm volatile("s_wait_asynccnt 0x3" ::: "memory");

        const float4 xv = ring[t & (NBUF - 1)][tid];   // ds_load_b128 (own bytes)
        u.x = __fmaf_rn(al.x, u.x, om.x * xv.x);
        u.y = __fmaf_rn(al.y, u.y, om.y * xv.y);
        u.z = __fmaf_rn(al.z, u.z, om.z * xv.z);
        u.w = __fmaf_rn(al.w, u.w, om.w * xv.w);
        o4[(size_t)t * COLS + c] = u;                  // global_store_b128
    }

    // ---- Tail: drain remaining NBUF-1 in-flight loads ----------------------
#pragma unroll
    for (int t = T_STEPS - (NBUF - 1); t < T_STEPS; ++t) {
        asm volatile("s_wait_asynccnt 0x0" ::: "memory");
        const float4 xv = ring[t & (NBUF - 1)][tid];
        u.x = __fmaf_rn(al.x, u.x, om.x * xv.x);
        u.y = __fmaf_rn(al.y, u.y, om.y * xv.y);
        u.z = __fmaf_rn(al.z, u.z, om.z * xv.z);
        u.w = __fmaf_rn(al.w, u.w, om.w * xv.w);
        o4[(size_t)t * COLS + c] = u;
    }
}

extern "C" void kernel_launch(void* const* d_in, const int* in_sizes, int n_in,
                              void* d_out, int out_size, void* d_ws, size_t ws_size,
                              hipStream_t stream) {
    (void)in_sizes; (void)n_in; (void)out_size; (void)d_ws; (void)ws_size;
    const float* x   = (const float*)d_in[0];   // [T*B, D] f32
    const float* tau = (const float*)d_in[1];   // [D] f32
    // d_in[2] is the Python scalar T (==128); baked in as a compile-time constant.
    float* out = (float*)d_out;

    dim3 grid(COLS / TPB);   // 256 blocks
    dim3 block(TPB);         // 256 threads (8 wave32s)
    li_scan_async<<<grid, block, 0, stream>>>(x, tau, out);
}